// ConditionDiffusion_90374701842591
// MI455X (gfx1250) — compile-verified
//
#include <hip/hip_runtime.h>
#include <hip/hip_bf16.h>
#include <stddef.h>

// ---------------------------------------------------------------------------
// Reverse-diffusion scan for MI455X (gfx1250), f32 WMMA path.
//   batch=512, steps=500, length=128, hidden=512
// Batch rows are independent -> 32 workgroups x 16 rows, each runs the full
// 499-step scan with per-step 16x512x136 and 16x128x512 f32 WMMA GEMMs.
// W1 fragments are register-resident across the step loop (1024-VGPR space);
// W2 fragments stream from the 192MB L2 via coalesced global_load_b128.
// ---------------------------------------------------------------------------

typedef float v2f __attribute__((ext_vector_type(2)));
typedef float v4f __attribute__((ext_vector_type(4)));
typedef float v8f __attribute__((ext_vector_type(8)));

#define BATCH   512
#define STEPS   500
#define LEN     128
#define HID     512
#define NPAIR1  17             // ceil(129/8) K-pairs for GEMM1 (K padded to 136)
#define NPAIR2  64             // 512/8     K-pairs for GEMM2
#define XS_STRIDE 140          // 140 % 64 == 12 -> conflict-free row skew
#define HS_STRIDE 516          // 516 % 64 == 4  -> conflict-free row skew
#define DT_F     0.001f
#define SQRT_DT  0.03162277660168379332f

// W1 fragments: [ntile(32)][kpair(17)][lane(32)][4 floats] = 69632 floats
// W2 fragments: [ntile(8)][kpair(64)][lane(32)][4 floats]  = 65536 floats
#define W1F_FLOATS (32 * NPAIR1 * 32 * 4)
#define W2F_FLOATS (8  * NPAIR2 * 32 * 4)

__device__ __forceinline__ v8f wmma4(v2f a, v2f b, v8f c) {
  // V_WMMA_F32_16X16X4_F32 : D = A(16x4) * B(4x16) + C(16x16), all f32
  return __builtin_amdgcn_wmma_f32_16x16x4_f32(
      /*neg_a=*/false, a, /*neg_b=*/false, b,
      /*c_mod=*/(short)0, c, /*reuse_a=*/false, /*reuse_b=*/false);
}

// ---------------------------------------------------------------------------
// One-time weight swizzle into per-lane WMMA B-fragment order so each lane
// fetches two K-steps of B with one coalesced global_load_b128.
// Fragment element (ntile, kpair p, lane, j):
//   n = ntile*16 + (lane&15)
//   k = 4*(2p + (j>>1)) + ((lane>>4)<<1) + (j&1)
// ---------------------------------------------------------------------------
__global__ __launch_bounds__(256) void swizzle_weights(
    const float* __restrict__ W1, const float* __restrict__ W2,
    float* __restrict__ W1f, float* __restrict__ W2f) {
  int gid = blockIdx.x * 256 + threadIdx.x;
  if (gid < W1F_FLOATS) {
    int j    = gid & 3;
    int lane = (gid >> 2) & 31;
    int rest = gid >> 7;           // nt*17 + p
    int p    = rest % NPAIR1;
    int nt   = rest / NPAIR1;
    int n    = nt * 16 + (lane & 15);
    int k    = 4 * (2 * p + (j >> 1)) + ((lane >> 4) << 1) + (j & 1);
    W1f[gid] = (k < LEN + 1) ? W1[k * HID + n] : 0.0f;
  } else if (gid < W1F_FLOATS + W2F_FLOATS) {
    int g    = gid - W1F_FLOATS;
    int j    = g & 3;
    int lane = (g >> 2) & 31;
    int rest = g >> 7;             // nt*64 + p
    int p    = rest & 63;
    int nt   = rest >> 6;
    int n    = nt * 16 + (lane & 15);
    int k    = 4 * (2 * p + (j >> 1)) + ((lane >> 4) << 1) + (j & 1);
    W2f[g]   = W2[k * LEN + n];
  }
}

// ---------------------------------------------------------------------------
// Persistent scan kernel: block b owns batch rows [16b, 16b+16).
// 8 waves: GEMM1 -> 4 hidden N-tiles/wave (4 independent acc chains),
//          GEMM2 -> 1 length N-tile/wave (4 independent acc chains).
// ---------------------------------------------------------------------------
__global__ __launch_bounds__(256) void diffusion_scan(
    const float* __restrict__ fd,     // (512, 500, 128)
    const float* __restrict__ b1,     // (512,)
    const float* __restrict__ b2,     // (128,)
    const float* __restrict__ noise,  // (499, 128)
    const float* __restrict__ W1f,
    const float* __restrict__ W2f,
    float* __restrict__ out) {        // (512, 500, 128)
  __shared__ float xs[16 * XS_STRIDE];   // [x | t | zero-pad] per row
  __shared__ float hs[16 * HS_STRIDE];   // relu hidden activations

  const int tid   = threadIdx.x;
  const int wave  = tid >> 5;
  const int lane  = tid & 31;
  const int lrow  = lane & 15;   // M row (A/C) or N col (B/C) within tile
  const int lhalf = lane >> 4;   // selects K {0,1} vs {2,3} / M row vs row+8
  const int r0    = blockIdx.x * 16;

  // ---- init xs tile (cols 0..127 = x0, 128 = t slot, 129..139 = 0) ----
  for (int idx = tid; idx < 16 * XS_STRIDE; idx += 256) {
    int m = idx / XS_STRIDE;
    int c = idx - m * XS_STRIDE;
    float v = 0.0f;
    if (c < LEN) v = fd[((size_t)(r0 + m) * STEPS + (STEPS - 1)) * LEN + c];
    xs[idx] = v;
  }
  // ---- out[:, 0, :] = x0 ----
  for (int idx = tid; idx < 16 * LEN; idx += 256) {
    int m = idx >> 7;
    int c = idx & (LEN - 1);
    out[((size_t)(r0 + m) * STEPS) * LEN + c] =
        fd[((size_t)(r0 + m) * STEPS + (STEPS - 1)) * LEN + c];
  }

  // ---- per-lane bias fragments (constant across all 8 C VGPRs) ----
  float bias1[4];
  for (int q = 0; q < 4; ++q) bias1[q] = b1[(wave * 4 + q) * 16 + lrow];
  const float bias2v = b2[wave * 16 + lrow];

  // ---- per-lane pointers ----
  const v2f* px = (const v2f*)(xs + lrow * XS_STRIDE + lhalf * 2);
  const v2f* ph = (const v2f*)(hs + lrow * HS_STRIDE + lhalf * 2);
  const v4f* pw1 = (const v4f*)W1f + (size_t)wave * 4 * NPAIR1 * 32 + lane;
  const v4f* pw2 = (const v4f*)W2f + (size_t)wave * NPAIR2 * 32 + lane;

  __syncthreads();

#pragma unroll 1
  for (int i = 0; i < STEPS - 1; ++i) {
    const float tcur = DT_F * (float)(STEPS - 1 - i);
    if (tid < 16) xs[tid * XS_STRIDE + LEN] = tcur;   // t column
    __syncthreads();

    // ================= GEMM1: h = relu([x,t] @ W1 + b1) =================
    // 4 N-tiles -> 4 independent accumulator chains; W1 fragments are
    // loop-invariant and stay register-resident (no global traffic).
    v8f acc[4];
#pragma unroll
    for (int q = 0; q < 4; ++q) {
      v8f a;
      for (int v = 0; v < 8; ++v) a[v] = bias1[q];
      acc[q] = a;
    }
#pragma unroll
    for (int p = 0; p < NPAIR1; ++p) {
      v2f a0 = px[4 * p];        // K-step 2p   (ds_load_2addr_b64)
      v2f a1 = px[4 * p + 2];    // K-step 2p+1
#pragma unroll
      for (int q = 0; q < 4; ++q) {
        v4f w = pw1[(q * NPAIR1 + p) * 32];
        v2f w0; w0.x = w.x; w0.y = w.y;
        v2f w1; w1.x = w.z; w1.y = w.w;
        acc[q] = wmma4(a0, w0, acc[q]);
        acc[q] = wmma4(a1, w1, acc[q]);
      }
    }
    // relu + stage h into LDS (row = v + 8*lhalf, col = ntile*16 + lrow)
#pragma unroll
    for (int q = 0; q < 4; ++q) {
      int n = (wave * 4 + q) * 16 + lrow;
#pragma unroll
      for (int v = 0; v < 8; ++v) {
        float h = acc[q][v];
        hs[(v + 8 * lhalf) * HS_STRIDE + n] = h > 0.0f ? h : 0.0f;
      }
    }
    __syncthreads();

    // ================= GEMM2: score = h @ W2 + b2 =================
    // Streamed from L2 (register pressure stays below spill threshold);
    // 4 independent accumulator chains (p mod 4); unroll 8 keeps 8 coalesced
    // b128 loads in flight to cover L2 latency.
    v8f sc[4];
    {
      v8f z;
      for (int v = 0; v < 8; ++v) z[v] = 0.0f;
      v8f s0;
      for (int v = 0; v < 8; ++v) s0[v] = bias2v;
      sc[0] = s0; sc[1] = z; sc[2] = z; sc[3] = z;
    }
#pragma unroll 8
    for (int p = 0; p < NPAIR2; ++p) {
      v2f a0 = ph[4 * p];
      v2f a1 = ph[4 * p + 2];
      v4f w = pw2[p * 32];               // global_load_b128, coalesced, L2-hot
      v2f w0; w0.x = w.x; w0.y = w.y;
      v2f w1; w1.x = w.z; w1.y = w.w;
      sc[p & 3] = wmma4(a0, w0, sc[p & 3]);
      sc[p & 3] = wmma4(a1, w1, sc[p & 3]);
    }
    v8f scf;
#pragma unroll
    for (int v = 0; v < 8; ++v)
      scf[v] = (sc[0][v] + sc[1][v]) + (sc[2][v] + sc[3][v]);

    // ================= SDE update + output store =================
    {
      int n = wave * 16 + lrow;            // exclusive column per wave
      float eps = noise[i * LEN + n];
      int po = STEPS - 1 - i;              // ys[i] lands at out[:, 499-i, :]
#pragma unroll
      for (int v = 0; v < 8; ++v) {
        int m = v + 8 * lhalf;
        float xv = xs[m * XS_STRIDE + n];
        float xn = xv + (-0.5f * xv - scf[v]) * DT_F + SQRT_DT * eps;
        xs[m * XS_STRIDE + n] = xn;
        out[((size_t)(r0 + m) * STEPS + po) * LEN + n] = xn;
      }
    }
    // next iteration's first __syncthreads() makes xs updates visible
  }
}

// ---------------------------------------------------------------------------
extern "C" void kernel_launch(void* const* d_in, const int* in_sizes, int n_in,
                              void* d_out, int out_size, void* d_ws, size_t ws_size,
                              hipStream_t stream) {
  const float* fd    = (const float*)d_in[0];   // forward_diffusion
  const float* W1    = (const float*)d_in[1];
  const float* b1    = (const float*)d_in[2];
  const float* W2    = (const float*)d_in[3];
  const float* b2    = (const float*)d_in[4];
  const float* noise = (const float*)d_in[5];
  float* out = (float*)d_out;

  float* W1f = (float*)d_ws;              // 69632 floats
  float* W2f = W1f + W1F_FLOATS;          // 65536 floats (total ~528 KB of ws)

  int total = W1F_FLOATS + W2F_FLOATS;
  swizzle_weights<<<(total + 255) / 256, 256, 0, stream>>>(W1, W2, W1f, W2f);

  diffusion_scan<<<BATCH / 16, 256, 0, stream>>>(fd, b1, b2, noise, W1f, W2f, out);
}